// PatchEmbed_5583457485374
// MI455X (gfx1250) — compile-verified
//
#include <hip/hip_runtime.h>

typedef __attribute__((ext_vector_type(2))) float v2f;
typedef __attribute__((ext_vector_type(8))) float v8f;

#define NBATCH 8
#define NPTS   16384
#define MC     2048
#define KNN    32
#define EMB    256
#define PATCH_PER_WG 16
#define NWG    ((NBATCH * MC) / PATCH_PER_WG)   // 1024 workgroups

// Row-major staging, padded row stride (stride mod 64 == 4):
// A-tile b64 reads hit all 64 banks exactly once; D-tile stores conflict-free.
#define H1_STRIDE 68      // 32 rows x (64 + 4)
#define H2_STRIDE 132     // 32 rows x (128 + 4)
#define W3_STRIDE 130     // 256 rows x (128 + 2)

#define H1_FLOATS (32 * H1_STRIDE)     //  2176 floats,   8704 B
#define H2_FLOATS (32 * H2_STRIDE)     //  4224 floats,  16896 B
#define W3_FLOATS (EMB * W3_STRIDE)    // 33280 floats, 133120 B
#define H2_OFF    H1_FLOATS
#define W3_OFF    (H1_FLOATS + H2_FLOATS)

static __device__ __forceinline__ v8f wmma_f32(v2f a, v2f b, v8f c) {
  // D = A(16x4) * B(4x16) + C, all f32.
  return __builtin_amdgcn_wmma_f32_16x16x4_f32(false, a, false, b, (short)0, c,
                                               false, false);
}

// Single-instruction ReLU: WMMA output is already canonical, skip the
// canonicalize max(v,v) the compiler inserts for fmaxf.
static __device__ __forceinline__ float relu1(float x) {
  float r;
  asm("v_max_num_f32 %0, 0, %1" : "=v"(r) : "v"(x));
  return r;
}

__global__ __launch_bounds__(256) void patch_embed_wmma(
    const float* __restrict__ xyz, const float* __restrict__ centers,
    const int* __restrict__ idx_knn,
    const float* __restrict__ W1, const float* __restrict__ b1,
    const float* __restrict__ W2, const float* __restrict__ b2,
    const float* __restrict__ W3, const float* __restrict__ b3,
    float* __restrict__ out)
{
  // Single merged LDS block so WE control the layout: hot h1/h2 live below the
  // 64KB DS-immediate window; big w3 staging sits at the end.
  __shared__ float smem[H1_FLOATS + H2_FLOATS + W3_FLOATS];  // 158720 B total
  float* const h1s = smem;
  float* const h2s = smem + H2_OFF;
  float* const w3s = smem + W3_OFF;

  const int tid  = threadIdx.x;
  const int lane = tid & 31;
  const int wave = tid >> 5;
  const int l15  = lane & 15;
  const int hi   = lane >> 4;              // 0: lanes 0-15 hold K0/K1; 1: lanes 16-31 hold K2/K3

  // ---- Stage W3 (256x128 row-major) into padded LDS rows, float2 granules ----
  for (int i = tid; i < EMB * 64; i += 256) {
    const int r = i >> 6;
    const int c = (i & 63) << 1;
    const float2 v = *reinterpret_cast<const float2*>(W3 + r * 128 + c);
    w3s[r * W3_STRIDE + c]     = v.x;
    w3s[r * W3_STRIDE + c + 1] = v.y;
  }

  // ---------------- Patch-invariant per-lane values (hoisted) ----------------
  // GEMM1: wave -> (mt1, nt1) tile; B tile of W1^T and bias.
  const int   mt1 = wave & 1;
  const int   n1  = (wave >> 1) * 16 + l15;
  const float w1x = W1[n1 * 3 + (hi << 1)];        // K0 (lo) / K2 (hi)
  const float w1yl = W1[n1 * 3 + 1];
  const float w1y = hi ? 0.f : w1yl;               // K1 (lo) / pad (hi)

  // GEMM2: wave -> N slice n2; hoisted W2 row pointer with hi-offset folded in.
  const int    n2    = wave * 16 + l15;
  const float* w2row = W2 + n2 * 64 + (hi << 1);

  // GEMM3: wave -> 2 N slices; hoisted LDS row pointers with hi-offset folded in.
  const int    nA = wave * 32 + l15;
  const int    nB = nA + 16;
  const float  biasA = b3[nA], biasB = b3[nB];
  const float* w3rowA = w3s + nA * W3_STRIDE + (hi << 1);
  const float* w3rowB = w3s + nB * W3_STRIDE + (hi << 1);

  // A-tile row pointers (rows l15 / 16+l15) with hi K-offset folded in.
  const float* h1r0 = h1s + l15 * H1_STRIDE + (hi << 1);   // +4352B imm for row+16
  const float* h2r0 = h2s + l15 * H2_STRIDE + (hi << 1);   // +8448B imm for row+16

  // D-tile store bases (column n fixed per lane, 8 consecutive padded rows).
  float* const h1st  = h1s + (mt1 * 16 + hi * 8) * H1_STRIDE + n1;
  float* const h2st0 = h2s + (hi * 8) * H2_STRIDE + n2;         // rows 0-15 tile
  float* const h2st1 = h2s + (16 + hi * 8) * H2_STRIDE + n2;    // rows 16-31 tile

  // Hoisted C-operand vectors: first WMMA of each chain reads these directly
  // (no per-patch accumulator-init movs).
  const float bias1v = b1[n1];
  const float bias2v = b2[n2];
  v8f cb1, cb2, czero;
  #pragma unroll
  for (int j = 0; j < 8; ++j) { cb1[j] = bias1v; cb2[j] = bias2v; czero[j] = 0.f; }

  __syncthreads();

  const int p0 = blockIdx.x * PATCH_PER_WG;
  for (int p = p0; p < p0 + PATCH_PER_WG; ++p) {
    const int b = p >> 11;                         // p / MC
    const float* ctr  = centers + p * 3;
    const int*   idxp = idx_knn + p * KNN;
    const float* xyzb = xyz + (size_t)b * NPTS * 3;

    // ============ GEMM1: local(32x3) @ W1^T(3x64) -> h1(32x64), one tile/wave ============
    {
      const int r  = mt1 * 16 + l15;
      const int nb = idxp[r];
      const float* q = xyzb + (size_t)nb * 3;
      const float lx = q[0] - ctr[0], ly = q[1] - ctr[1], lz = q[2] - ctr[2];
      v2f a;  a.x = hi ? lz : lx;  a.y = hi ? 0.f : ly;   // K=3 padded to 4
      v2f bm; bm.x = w1x; bm.y = w1y;
      const v8f d = wmma_f32(a, bm, cb1);
      #pragma unroll
      for (int j = 0; j < 8; ++j)
        h1st[j * H1_STRIDE] = relu1(d[j]);
    }
    __syncthreads();

    // ============ GEMM2: h1(32x64) @ W2^T(64x128) -> h2(32x128), N-slice/wave ============
    {
      v8f acc0 = cb2, acc1 = cb2;
      #pragma unroll
      for (int ks = 0; ks < 16; ++ks) {
        const float2 a0v = *reinterpret_cast<const float2*>(h1r0 + ks * 4);
        const float2 a1v = *reinterpret_cast<const float2*>(h1r0 + 16 * H1_STRIDE + ks * 4);
        const float2 wv  = *reinterpret_cast<const float2*>(w2row + ks * 4);
        v2f a0; a0.x = a0v.x; a0.y = a0v.y;
        v2f a1; a1.x = a1v.x; a1.y = a1v.y;
        v2f bm; bm.x = wv.x;  bm.y = wv.y;
        acc0 = wmma_f32(a0, bm, acc0);
        acc1 = wmma_f32(a1, bm, acc1);
      }
      #pragma unroll
      for (int j = 0; j < 8; ++j) {
        h2st0[j * H2_STRIDE] = relu1(acc0[j]);
        h2st1[j * H2_STRIDE] = relu1(acc1[j]);
      }
    }
    __syncthreads();

    // ======== GEMM3: h2(32x128) @ W3^T(128x256) + max over 32 rows, 2 N-slices/wave ========
    {
      // Bias deferred past the max-pool (per-column bias, max over rows).
      v8f aA0 = czero, aA1 = czero, aB0 = czero, aB1 = czero;
      #pragma unroll
      for (int ks = 0; ks < 32; ++ks) {
        const float2 a0v = *reinterpret_cast<const float2*>(h2r0 + ks * 4);
        const float2 a1v = *reinterpret_cast<const float2*>(h2r0 + 16 * H2_STRIDE + ks * 4);
        const float2 wAv = *reinterpret_cast<const float2*>(w3rowA + ks * 4);
        const float2 wBv = *reinterpret_cast<const float2*>(w3rowB + ks * 4);
        v2f a0; a0.x = a0v.x; a0.y = a0v.y;
        v2f a1; a1.x = a1v.x; a1.y = a1v.y;
        v2f bA; bA.x = wAv.x; bA.y = wAv.y;
        v2f bB; bB.x = wBv.x; bB.y = wBv.y;
        aA0 = wmma_f32(a0, bA, aA0);
        aA1 = wmma_f32(a1, bA, aA1);
        aB0 = wmma_f32(a0, bB, aB0);
        aB1 = wmma_f32(a1, bB, aB1);
      }
      // Max over K=32 rows: 8 D-VGPRs x 2 M-tiles in-lane, then half-wave swap.
      float mA = fmaxf(aA0[0], aA1[0]);
      float mB = fmaxf(aB0[0], aB1[0]);
      #pragma unroll
      for (int j = 1; j < 8; ++j) {
        mA = fmaxf(mA, fmaxf(aA0[j], aA1[j]));
        mB = fmaxf(mB, fmaxf(aB0[j], aB1[j]));
      }
      mA = fmaxf(mA, __shfl_xor(mA, 16, 32)) + biasA;
      mB = fmaxf(mB, __shfl_xor(mB, 16, 32)) + biasB;
      // All lanes hold both pooled values; lane l stores column wave*32 + l.
      out[(size_t)p * EMB + wave * 32 + lane] = (lane < 16) ? mA : mB;
    }
  }
}

extern "C" void kernel_launch(void* const* d_in, const int* in_sizes, int n_in,
                              void* d_out, int out_size, void* d_ws, size_t ws_size,
                              hipStream_t stream) {
  const float* xyz     = (const float*)d_in[0];
  const float* centers = (const float*)d_in[1];
  const int*   idx     = (const int*)  d_in[2];
  const float* W1      = (const float*)d_in[3];
  const float* b1      = (const float*)d_in[4];
  const float* W2      = (const float*)d_in[5];
  const float* b2      = (const float*)d_in[6];
  const float* W3      = (const float*)d_in[7];
  const float* b3      = (const float*)d_in[8];
  float* out = (float*)d_out;

  dim3 grid(NWG), block(256);
  hipLaunchKernelGGL(patch_embed_wmma, grid, block, 0, stream,
                     xyz, centers, idx, W1, b1, W2, b2, W3, b3, out);
}